// GCN_Pyg_23347442221163
// MI455X (gfx1250) — compile-verified
//
#include <hip/hip_runtime.h>
#include <hip/hip_bf16.h>

typedef float v2f __attribute__((ext_vector_type(2)));
typedef float v8f __attribute__((ext_vector_type(8)));

#define NNODES   50000
#define NEDGES   1600000
#define C_IN     128
#define C_HID    128
#define C_OUT    10
#define C_OUTP   16   // padded

// ---------------------------------------------------------------------------
// deg[dst] += ew  (one thread per edge)
// ---------------------------------------------------------------------------
__global__ __launch_bounds__(256) void edge_deg_kernel(
    const long long* __restrict__ dst, const float* __restrict__ ew,
    float* __restrict__ deg, int E)
{
    int e = blockIdx.x * blockDim.x + threadIdx.x;
    if (e >= E) return;
    atomicAdd(&deg[dst[e]], ew[e]);
}

// deg -> dinv = rsqrt(deg + 1) in place (self-loop adds 1, always > 0)
__global__ __launch_bounds__(256) void node_dinv_kernel(float* __restrict__ deg, int n)
{
    int i = blockIdx.x * blockDim.x + threadIdx.x;
    if (i >= n) return;
    deg[i] = rsqrtf(deg[i] + 1.0f);
}

// ---------------------------------------------------------------------------
// xw1 = x @ W1   (N x 128) = (N x 128)(128 x 128), fp32 WMMA 16x16x4
// 8 waves / block; W1 staged in 64KB LDS in K-pair-interleaved float2 layout:
//   sW[p*128 + n] = { W[2p][n], W[2p+1][n] }
// so every B fragment is one 8-byte ds_load_b64 straight into a VGPR pair.
// ---------------------------------------------------------------------------
__global__ __launch_bounds__(256) void gemm_x_w1_kernel(
    const float* __restrict__ x, const float* __restrict__ W,
    float* __restrict__ out, int nrows)
{
    __shared__ float2 sW[(C_IN / 2) * C_HID];       // 64 KB (WGP has 320 KB)
    for (int i = threadIdx.x; i < (C_IN / 2) * C_HID; i += blockDim.x) {
        int p = i >> 7, n = i & 127;                // pair index, column
        float2 v;
        v.x = W[(2 * p)     * C_HID + n];
        v.y = W[(2 * p + 1) * C_HID + n];
        sW[i] = v;
    }
    __syncthreads();

    const int wave = threadIdx.x >> 5;
    const int lane = threadIdx.x & 31;
    const int m0   = blockIdx.x * 128 + wave * 16;
    if (m0 >= nrows) return;                        // nrows % 16 == 0 -> full tiles

    const int mrow = lane & 15;                     // A: M index
    const int kh   = lane >> 4;                     // K-half (0/1) & C row-half
    const int ncol = lane & 15;                     // B/C/D: N index

    v8f acc[8] = {};                                // 8 column tiles of 16

    const float2* xrow2 = (const float2*)(x + (size_t)(m0 + mrow) * C_IN);

    for (int k0 = 0; k0 < C_IN; k0 += 4) {
        // A 16x4 f32: vgpr0 = K = 2*kh, vgpr1 = K = 2*kh+1 (8B-aligned pair)
        float2 af = xrow2[(k0 >> 1) + kh];
        v2f a; a.x = af.x; a.y = af.y;
        // B pair for this K-step: pair index p = k0/2 + kh
        const v2f* bbase = (const v2f*)&sW[((k0 >> 1) + kh) * C_HID + ncol];
        #pragma unroll
        for (int j = 0; j < 8; ++j) {
            v2f b = bbase[j * 16];                  // single ds_load_b64
            acc[j] = __builtin_amdgcn_wmma_f32_16x16x4_f32(
                false, a, false, b, (short)0, acc[j], false, false);
        }
    }

    // C/D layout: vgpr r holds M = r + 8*kh, N = ncol
    #pragma unroll
    for (int j = 0; j < 8; ++j) {
        float* orow = out + (size_t)(m0 + 8 * kh) * C_HID + j * 16 + ncol;
        #pragma unroll
        for (int r = 0; r < 8; ++r)
            orow[(size_t)r * C_HID] = acc[j][r];
    }
}

// ---------------------------------------------------------------------------
// xw2 = h @ W2pad  (N x 16) = (N x 128)(128 x 16), W2 zero-padded 10 -> 16,
// staged in LDS with the same K-pair-interleaved float2 layout.
// ---------------------------------------------------------------------------
__global__ __launch_bounds__(256) void gemm_h_w2_kernel(
    const float* __restrict__ h, const float* __restrict__ W2,
    float* __restrict__ out, int nrows)
{
    __shared__ float2 sW[(C_HID / 2) * C_OUTP];     // 8 KB
    for (int i = threadIdx.x; i < (C_HID / 2) * C_OUTP; i += blockDim.x) {
        int p = i >> 4, c = i & 15;
        float2 v;
        v.x = (c < C_OUT) ? W2[(2 * p)     * C_OUT + c] : 0.0f;
        v.y = (c < C_OUT) ? W2[(2 * p + 1) * C_OUT + c] : 0.0f;
        sW[i] = v;
    }
    __syncthreads();

    const int wave = threadIdx.x >> 5;
    const int lane = threadIdx.x & 31;
    const int m0   = blockIdx.x * 128 + wave * 16;
    if (m0 >= nrows) return;

    const int mrow = lane & 15;
    const int kh   = lane >> 4;
    const int ncol = lane & 15;

    v8f acc = {};
    const float2* hrow2 = (const float2*)(h + (size_t)(m0 + mrow) * C_HID);

    for (int k0 = 0; k0 < C_HID; k0 += 4) {
        float2 af = hrow2[(k0 >> 1) + kh];
        v2f a; a.x = af.x; a.y = af.y;
        v2f b = *(const v2f*)&sW[((k0 >> 1) + kh) * C_OUTP + ncol];
        acc = __builtin_amdgcn_wmma_f32_16x16x4_f32(
            false, a, false, b, (short)0, acc, false, false);
    }

    #pragma unroll
    for (int r = 0; r < 8; ++r)
        out[(size_t)(m0 + 8 * kh + r) * C_OUTP + ncol] = acc[r];
}

// ---------------------------------------------------------------------------
// Layer-1 edge aggregation: agg[dst] += xw[src] * norm, 128 channels.
// One wave32 per edge; lane owns 4 channels (float4 gather + 4 f32 atomics).
// xw/agg are 25.6 MB each -> resident in the 192 MB L2.
// ---------------------------------------------------------------------------
__global__ __launch_bounds__(256) void edge_agg128_kernel(
    const long long* __restrict__ src, const long long* __restrict__ dst,
    const float* __restrict__ ew, const float* __restrict__ dinv,
    const float* __restrict__ xw, float* __restrict__ agg, int E)
{
    int e = blockIdx.x * 8 + (threadIdx.x >> 5);
    if (e >= E) return;
    int lane = threadIdx.x & 31;
    long long s = src[e], d = dst[e];
    float norm = dinv[s] * ew[e] * dinv[d];
    float4 v = ((const float4*)(xw + (size_t)s * C_HID))[lane];
    float* ap = agg + (size_t)d * C_HID + lane * 4;
    atomicAdd(ap + 0, v.x * norm);
    atomicAdd(ap + 1, v.y * norm);
    atomicAdd(ap + 2, v.z * norm);
    atomicAdd(ap + 3, v.w * norm);
}

// Layer-2 edge aggregation: 16 channels (10 live); 2 edges per wave.
__global__ __launch_bounds__(256) void edge_agg16_kernel(
    const long long* __restrict__ src, const long long* __restrict__ dst,
    const float* __restrict__ ew, const float* __restrict__ dinv,
    const float* __restrict__ xw, float* __restrict__ agg, int E)
{
    int wid  = blockIdx.x * 8 + (threadIdx.x >> 5);
    int lane = threadIdx.x & 31;
    int e = wid * 2 + (lane >> 4);
    int c = lane & 15;
    if (e >= E || c >= C_OUT) return;
    long long s = src[e], d = dst[e];
    float norm = dinv[s] * ew[e] * dinv[d];
    atomicAdd(&agg[(size_t)d * C_OUTP + c], xw[(size_t)s * C_OUTP + c] * norm);
}

// ---------------------------------------------------------------------------
// h = relu(agg1 + xw1 * dinv^2 + b1), written in place into agg1
// ---------------------------------------------------------------------------
__global__ __launch_bounds__(256) void selfloop_relu_kernel(
    const float* __restrict__ xw, const float* __restrict__ dinv,
    const float* __restrict__ b, float* __restrict__ agg, int total)
{
    int i = blockIdx.x * blockDim.x + threadIdx.x;
    if (i >= total) return;
    int n = i >> 7, c = i & 127;
    float di = dinv[n];
    float v = agg[i] + xw[i] * (di * di) + b[c];
    agg[i] = fmaxf(v, 0.0f);
}

// ---------------------------------------------------------------------------
// out = log_softmax(agg2 + xw2 * dinv^2 + b2), one thread per node (10 ch)
// ---------------------------------------------------------------------------
__global__ __launch_bounds__(256) void final_logsoftmax_kernel(
    const float* __restrict__ xw2, const float* __restrict__ agg2,
    const float* __restrict__ dinv, const float* __restrict__ b2,
    float* __restrict__ out, int n)
{
    int i = blockIdx.x * blockDim.x + threadIdx.x;
    if (i >= n) return;
    float di2 = dinv[i] * dinv[i];
    float o[C_OUT];
    float mx = -__builtin_inff();
    #pragma unroll
    for (int c = 0; c < C_OUT; ++c) {
        float v = agg2[(size_t)i * C_OUTP + c] + xw2[(size_t)i * C_OUTP + c] * di2 + b2[c];
        o[c] = v;
        mx = fmaxf(mx, v);
    }
    float sum = 0.0f;
    #pragma unroll
    for (int c = 0; c < C_OUT; ++c) sum += __expf(o[c] - mx);
    float lse = mx + __logf(sum);
    #pragma unroll
    for (int c = 0; c < C_OUT; ++c) out[(size_t)i * C_OUT + c] = o[c] - lse;
}

// ---------------------------------------------------------------------------
extern "C" void kernel_launch(void* const* d_in, const int* in_sizes, int n_in,
                              void* d_out, int out_size, void* d_ws, size_t ws_size,
                              hipStream_t stream)
{
    const float*     x   = (const float*)d_in[0];
    const long long* adj = (const long long*)d_in[1];   // int64 [2, E]
    const float*     ew  = (const float*)d_in[2];
    const float*     W1  = (const float*)d_in[3];
    const float*     b1  = (const float*)d_in[4];
    const float*     W2  = (const float*)d_in[5];
    const float*     b2  = (const float*)d_in[6];
    float*           out = (float*)d_out;

    const int N = NNODES;
    const int E = NEDGES;
    const long long* srcIdx = adj;
    const long long* dstIdx = adj + E;

    // Workspace layout (floats): dinv | xw1 | agg1(->h) | xw2 | agg2  (~58 MB)
    float* ws   = (float*)d_ws;
    float* dinv = ws;
    float* xw1  = dinv + N;
    float* agg1 = xw1 + (size_t)N * C_HID;
    float* xw2  = agg1 + (size_t)N * C_HID;
    float* agg2 = xw2 + (size_t)N * C_OUTP;

    hipMemsetAsync(dinv, 0, (size_t)N * sizeof(float), stream);
    hipMemsetAsync(agg1, 0, (size_t)N * C_HID * sizeof(float), stream);
    hipMemsetAsync(agg2, 0, (size_t)N * C_OUTP * sizeof(float), stream);

    edge_deg_kernel<<<(E + 255) / 256, 256, 0, stream>>>(dstIdx, ew, dinv, E);
    node_dinv_kernel<<<(N + 255) / 256, 256, 0, stream>>>(dinv, N);

    gemm_x_w1_kernel<<<(N + 127) / 128, 256, 0, stream>>>(x, W1, xw1, N);
    edge_agg128_kernel<<<(E + 7) / 8, 256, 0, stream>>>(srcIdx, dstIdx, ew, dinv, xw1, agg1, E);
    selfloop_relu_kernel<<<(N * C_HID + 255) / 256, 256, 0, stream>>>(xw1, dinv, b1, agg1, N * C_HID);

    gemm_h_w2_kernel<<<(N + 127) / 128, 256, 0, stream>>>(agg1, W2, xw2, N);
    edge_agg16_kernel<<<(E + 15) / 16, 256, 0, stream>>>(srcIdx, dstIdx, ew, dinv, xw2, agg2, E);
    final_logsoftmax_kernel<<<(N + 255) / 256, 256, 0, stream>>>(xw2, agg2, dinv, b2, out, N);
}